// LRNNCell_22076131901856
// MI455X (gfx1250) — compile-verified
//
#include <hip/hip_runtime.h>
#include <hip/hip_bf16.h>
#include <math.h>

// ---------------------------------------------------------------------------
// LRNN cell for MI455X (gfx1250, wave32).
//   Phase 0: transpose weights to ws (fp32), copy h0, reset grid barrier.
//   Phase 1: xp = x @ Whx + bhx  via V_WMMA_F32_16X16X4_F32 (32x32 tile per
//            wave: 4 WMMAs per 2 A-frag + 2 B-frag loads -> 1 load/WMMA),
//            written straight into the y-region of d_out.
//   Phase 2: persistent scan kernel, 512 steps, one grid barrier per step:
//            h' = tanh(xp_t + h @ Whh);  y_t = tanh(h' @ Wyh + byh)
// fp32 WMMA everywhere: exact numerics vs. the fp32 reference; the scan is
// sync-latency bound so low-precision WMMA would buy nothing there, and the
// big GEMM at fp32-WMMA rate is small vs. the 512 serialized steps.
// ---------------------------------------------------------------------------

#define B_   64
#define T_   512
#define NX_  1024
#define NH_  1024

#define SCAN_GRID   64
#define SCAN_BLOCK  128   // 4 waves -> 256 waves total = 256 tiles of 16x16

typedef __attribute__((ext_vector_type(2))) float v2f;
typedef __attribute__((ext_vector_type(8))) float v8f;

// ---------------------------- prep kernels ---------------------------------

__global__ void transpose1024_kernel(const float* __restrict__ src,
                                     float* __restrict__ dst) {
    __shared__ float tile[32][33];
    const int bx = blockIdx.x * 32, by = blockIdx.y * 32;
    const int tx = threadIdx.x, ty = threadIdx.y;   // block (32, 8)
    #pragma unroll
    for (int i = 0; i < 32; i += 8)
        tile[ty + i][tx] = src[(size_t)(by + ty + i) * 1024 + bx + tx];
    __syncthreads();
    #pragma unroll
    for (int i = 0; i < 32; i += 8)
        dst[(size_t)(bx + ty + i) * 1024 + by + tx] = tile[tx][ty + i];
}

__global__ void init_kernel(const float* __restrict__ h0,
                            float* __restrict__ hbuf0,
                            int* __restrict__ bar) {
    const int i = blockIdx.x * blockDim.x + threadIdx.x;
    if (i < B_ * NH_) hbuf0[i] = h0[i];
    if (i == 0) { bar[0] = 0; bar[1] = 0; }
}

// ------------------------- big GEMM: xp = x@Whx + b -------------------------
// One wave per 32x32 output tile: 4 fp32-WMMA accumulators, A/B fragment
// reuse gives 1 global load per WMMA instead of 2.

__global__ void gemm_xp_kernel(const float* __restrict__ x,     // [B*T, NX]
                               const float* __restrict__ WhxT,  // [NH, NX]
                               const float* __restrict__ bhx,   // [NH]
                               float* __restrict__ xp) {        // [B*T, NH]
    const int lane  = threadIdx.x & 31;
    const int wave  = threadIdx.x >> 5;
    const int lmod  = lane & 15;
    const int lhalf = lane >> 4;
    const int w     = blockIdx.x * (blockDim.x >> 5) + wave; // 0..32767
    const int tm    = w >> 5;                                // 32-row tile 0..1023
    const int tn    = w & 31;                                // 32-col tile 0..31

    const int row0 = tm * 32 + lmod;
    const int col0 = tn * 32 + lmod;

    const v2f* ap0 = (const v2f*)(x    + (size_t)row0 * NX_)        + lhalf;
    const v2f* ap1 = (const v2f*)(x    + (size_t)(row0 + 16) * NX_) + lhalf;
    const v2f* bp0 = (const v2f*)(WhxT + (size_t)col0 * NX_)        + lhalf;
    const v2f* bp1 = (const v2f*)(WhxT + (size_t)(col0 + 16) * NX_) + lhalf;

    v8f c00 = {}, c01 = {}, c10 = {}, c11 = {};
    #pragma unroll 4
    for (int i = 0; i < NX_ / 4; ++i) {
        const v2f a0 = ap0[2 * i];
        const v2f a1 = ap1[2 * i];
        const v2f b0 = bp0[2 * i];
        const v2f b1 = bp1[2 * i];
        c00 = __builtin_amdgcn_wmma_f32_16x16x4_f32(false, a0, false, b0,
                                                    (short)0, c00, false, false);
        c01 = __builtin_amdgcn_wmma_f32_16x16x4_f32(false, a0, false, b1,
                                                    (short)0, c01, false, false);
        c10 = __builtin_amdgcn_wmma_f32_16x16x4_f32(false, a1, false, b0,
                                                    (short)0, c10, false, false);
        c11 = __builtin_amdgcn_wmma_f32_16x16x4_f32(false, a1, false, b1,
                                                    (short)0, c11, false, false);
    }

    const float bias0 = bhx[col0];
    const float bias1 = bhx[col0 + 16];
    #pragma unroll
    for (int j = 0; j < 8; ++j) {
        const int m  = j + 8 * lhalf;                    // 0..15
        const size_t r0 = (size_t)(tm * 32 + m) * NH_;
        const size_t r1 = (size_t)(tm * 32 + 16 + m) * NH_;
        xp[r0 + col0]      = c00[j] + bias0;
        xp[r0 + col0 + 16] = c01[j] + bias1;
        xp[r1 + col0]      = c10[j] + bias0;
        xp[r1 + col0 + 16] = c11[j] + bias1;
    }
}

// ------------------------------ grid barrier --------------------------------

__device__ __forceinline__ void grid_sync(int* bar, int nblk) {
    __threadfence();            // make this block's global stores visible
    __syncthreads();
    if (threadIdx.x == 0) {
        const int g = __hip_atomic_load(&bar[1], __ATOMIC_RELAXED,
                                        __HIP_MEMORY_SCOPE_AGENT);
        const int a = __hip_atomic_fetch_add(&bar[0], 1, __ATOMIC_ACQ_REL,
                                             __HIP_MEMORY_SCOPE_AGENT);
        if (a == nblk - 1) {
            __hip_atomic_store(&bar[0], 0, __ATOMIC_RELAXED,
                               __HIP_MEMORY_SCOPE_AGENT);
            __hip_atomic_fetch_add(&bar[1], 1, __ATOMIC_RELEASE,
                                   __HIP_MEMORY_SCOPE_AGENT);
        } else {
            while (__hip_atomic_load(&bar[1], __ATOMIC_ACQUIRE,
                                     __HIP_MEMORY_SCOPE_AGENT) == g) {
                __builtin_amdgcn_s_sleep(2);
            }
        }
    }
    __syncthreads();
    __threadfence();
}

// ------------------------------ persistent scan -----------------------------
// 64 WGs x 4 waves = 256 waves; wave owns one 16x16 tile of the [64,1024]
// state. Per step: phase1 h' = tanh(xp_t + h@Whh) (double-buffered h),
// grid barrier, phase2 y_t = tanh(h'@Wyh + byh) overwriting xp_t in d_out.

__global__ void rnn_scan_kernel(const float* __restrict__ WhhT, // [NH, NH]
                                const float* __restrict__ WyhT, // [NH, NH]
                                const float* __restrict__ byh,  // [NH]
                                float* __restrict__ yx,         // d_out y region: xp in, y out
                                float* __restrict__ hbuf0,
                                float* __restrict__ hbuf1,
                                float* __restrict__ hout,       // d_out h_final
                                int* __restrict__ bar) {
    const int lane  = threadIdx.x & 31;
    const int wave  = threadIdx.x >> 5;
    const int lmod  = lane & 15;
    const int lhalf = lane >> 4;
    const int tile  = blockIdx.x * (SCAN_BLOCK / 32) + wave;  // 0..255
    const int tm    = tile >> 6;   // 0..3   (batch-row tile)
    const int tn    = tile & 63;   // 0..63  (column tile)
    const int rowA  = tm * 16 + lmod;   // batch row this lane loads for A
    const int col   = tn * 16 + lmod;   // output column this lane owns

    const v2f* bhh = (const v2f*)(WhhT + (size_t)col * NH_) + lhalf;
    const v2f* byw = (const v2f*)(WyhT + (size_t)col * NH_) + lhalf;
    const float bias_y = byh[col];

    for (int t = 0; t < T_; ++t) {
        const float* hcur = (t & 1) ? hbuf1 : hbuf0;
        float*       hnxt = (t & 1) ? hbuf0 : hbuf1;

        // ---- phase 1: h' = tanh(xp_t + hcur @ Whh)
        {
            const v2f* ah = (const v2f*)(hcur + (size_t)rowA * NH_) + lhalf;
            v8f c = {};
            #pragma unroll 8
            for (int i = 0; i < NH_ / 4; ++i) {
                c = __builtin_amdgcn_wmma_f32_16x16x4_f32(
                        false, ah[2 * i], false, bhh[2 * i], (short)0, c,
                        false, false);
            }
            #pragma unroll
            for (int j = 0; j < 8; ++j) {
                const int m = j + 8 * lhalf;          // 0..15
                const int b = tm * 16 + m;            // batch row
                const float xpv = yx[(size_t)(b * T_ + t) * NH_ + col];
                hnxt[(size_t)b * NH_ + col] = tanhf(c[j] + xpv);
            }
        }

        grid_sync(bar, SCAN_GRID);   // h' complete + visible device-wide

        // ---- phase 2: y_t = tanh(h' @ Wyh + byh), overwrite xp_t slot
        {
            const v2f* ay = (const v2f*)(hnxt + (size_t)rowA * NH_) + lhalf;
            v8f c = {};
            #pragma unroll 8
            for (int i = 0; i < NH_ / 4; ++i) {
                c = __builtin_amdgcn_wmma_f32_16x16x4_f32(
                        false, ay[2 * i], false, byw[2 * i], (short)0, c,
                        false, false);
            }
            #pragma unroll
            for (int j = 0; j < 8; ++j) {
                const int m = j + 8 * lhalf;
                const int b = tm * 16 + m;
                yx[(size_t)(b * T_ + t) * NH_ + col] = tanhf(c[j] + bias_y);
            }
        }
        // No second barrier needed: next step's phase 1 reads hnxt (guarded by
        // the barrier above) and writes the *other* h buffer.
    }

    // h_final lives in hbuf0 (T even); written before the last barrier.
    const float* hfin = (T_ & 1) ? hbuf1 : hbuf0;
    for (int i = blockIdx.x * blockDim.x + threadIdx.x; i < B_ * NH_;
         i += gridDim.x * blockDim.x)
        hout[i] = hfin[i];
}

// ------------------------------- launcher -----------------------------------

extern "C" void kernel_launch(void* const* d_in, const int* in_sizes, int n_in,
                              void* d_out, int out_size, void* d_ws, size_t ws_size,
                              hipStream_t stream) {
    const float* h   = (const float*)d_in[0];  // [B, NH]
    const float* x   = (const float*)d_in[1];  // [B, T, NX]
    const float* Whx = (const float*)d_in[2];  // [NX, NH]
    const float* bhx = (const float*)d_in[3];  // [NH]
    const float* Whh = (const float*)d_in[4];  // [NH, NH]
    const float* Wyh = (const float*)d_in[5];  // [NH, NH]
    const float* byh = (const float*)d_in[6];  // [NH]

    float* out = (float*)d_out;
    float* yx  = out + (size_t)B_ * NH_;       // y region, reused as xp scratch

    // workspace layout (floats): WhxT | WhhT | WyhT | hbuf0 | hbuf1 | barrier
    float* ws    = (float*)d_ws;
    float* WhxT  = ws;
    float* WhhT  = WhxT + (size_t)NH_ * NX_;
    float* WyhT  = WhhT + (size_t)NH_ * NH_;
    float* hbuf0 = WyhT + (size_t)NH_ * NH_;
    float* hbuf1 = hbuf0 + (size_t)B_ * NH_;
    int*   bar   = (int*)(hbuf1 + (size_t)B_ * NH_);

    // phase 0: transposes + init
    {
        dim3 blk(32, 8), grd(1024 / 32, 1024 / 32);
        transpose1024_kernel<<<grd, blk, 0, stream>>>(Whx, WhxT);
        transpose1024_kernel<<<grd, blk, 0, stream>>>(Whh, WhhT);
        transpose1024_kernel<<<grd, blk, 0, stream>>>(Wyh, WyhT);
        init_kernel<<<(B_ * NH_ + 255) / 256, 256, 0, stream>>>(h, hbuf0, bar);
    }

    // phase 1: xp = x @ Whx + bhx  -> written into y region of d_out
    {
        // 32x32 tiles: (B*T/32) * (NH/32) = 1024*32 = 32768 waves; 8 per block
        const int nblocks = (B_ * T_ / 32) * (NH_ / 32) / 8;
        gemm_xp_kernel<<<nblocks, 256, 0, stream>>>(x, WhxT, bhx, yx);
    }

    // phase 2: persistent recurrent scan
    rnn_scan_kernel<<<SCAN_GRID, SCAN_BLOCK, 0, stream>>>(
        WhhT, WyhT, byh, yx, hbuf0, hbuf1, out, bar);
}